// GraphConvolution_89094801588866
// MI455X (gfx1250) — compile-verified
//
#include <hip/hip_runtime.h>

typedef __attribute__((ext_vector_type(2))) float v2f;
typedef __attribute__((ext_vector_type(8))) float v8f;

#define IN_DIM  256
#define OUT_DIM 128
#define EDGES_PER_WAVE 16

// ---------------------------------------------------------------------------
// Kernel 1: HW = H @ W  via V_WMMA_F32_16X16X4_F32
// Grid: n_nodes/16 blocks x 256 threads (8 waves). Wave w computes the
// 16x16 tile at (blockIdx.x*16, w*16).
// A 16x4 f32 layout: lanes 0-15 hold M=0..15; v0 = K=0 (lo half) / K=2 (hi),
// v1 = K=1 / K=3.  B 4x16 mirrored: v0 = row K=0 / K=2, v1 = row K=1 / K=3.
// C/D: VGPR j -> M = j (lanes 0-15), M = 8+j (lanes 16-31), N = lane&15.
// ---------------------------------------------------------------------------
__global__ __launch_bounds__(256) void gemm_hw(const float* __restrict__ H,
                                               const float* __restrict__ W,
                                               float* __restrict__ HWout) {
    const int lane   = threadIdx.x & 31;
    const int waveId = threadIdx.x >> 5;
    const int m0     = blockIdx.x * 16;
    const int n0     = waveId * 16;

    const int row = lane & 15;          // M (for A) / N (for B)
    const int kb  = (lane >> 4) << 1;   // 0 for lanes 0-15, 2 for lanes 16-31

    const float* aptr = H + (m0 + row) * IN_DIM + kb;
    const float* bptr = W + kb * OUT_DIM + n0 + row;

    v8f acc = {};
#pragma unroll 4
    for (int k = 0; k < IN_DIM; k += 4) {
        v2f a = *(const v2f*)(aptr + k);          // H[m0+row][k+kb], [k+kb+1]
        v2f b;
        b.x = bptr[k * OUT_DIM];                  // W[k+kb  ][n0+row]
        b.y = bptr[k * OUT_DIM + OUT_DIM];        // W[k+kb+1][n0+row]
        acc = __builtin_amdgcn_wmma_f32_16x16x4_f32(
            /*neg_a=*/false, a, /*neg_b=*/false, b,
            /*c_mod=*/(short)0, acc, /*reuse_a=*/false, /*reuse_b=*/false);
    }

    const int mBase = m0 + ((lane >> 4) << 3);    // +8 for upper half-lanes
    const int col   = lane & 15;
    float* o = HWout + mBase * OUT_DIM + n0 + col;
#pragma unroll
    for (int j = 0; j < 8; ++j)
        o[j * OUT_DIM] = acc[j];
}

// ---------------------------------------------------------------------------
// Kernel 2: out[n][d] = bias[d]   (also re-initializes poisoned d_out)
// ---------------------------------------------------------------------------
__global__ void bias_init(const float* __restrict__ bias,
                          float* __restrict__ out, int total) {
    int i = blockIdx.x * blockDim.x + threadIdx.x;
    if (i < total) out[i] = bias[i & (OUT_DIM - 1)];
}

// ---------------------------------------------------------------------------
// Kernel 3: segmented COO SpMM scatter.  dst is sorted, so accumulate runs of
// equal dst in registers and flush with global_atomic_add_f32 on boundaries.
// One wave per EDGES_PER_WAVE contiguous edges; lane owns dims [4*lane,4*lane+4).
// ---------------------------------------------------------------------------
__global__ __launch_bounds__(256) void spmm_seg(const float* __restrict__ HW,
                                                const float* __restrict__ vals,
                                                const int*   __restrict__ src,
                                                const int*   __restrict__ dst,
                                                float* out, int E) {
    const int gwave = (int)((blockIdx.x * blockDim.x + threadIdx.x) >> 5);
    const int lane  = threadIdx.x & 31;
    const int e0    = gwave * EDGES_PER_WAVE;
    if (e0 >= E) return;
    const int e1 = min(e0 + EDGES_PER_WAVE, E);
    const int dbase = lane << 2;

    float4 acc = make_float4(0.f, 0.f, 0.f, 0.f);
    int cur = dst[e0];                     // uniform across the wave

    for (int e = e0; e < e1; ++e) {
        int d = dst[e];
        if (d != cur) {                    // uniform branch: flush run
            float* o = out + cur * OUT_DIM + dbase;
            unsafeAtomicAdd(o + 0, acc.x);
            unsafeAtomicAdd(o + 1, acc.y);
            unsafeAtomicAdd(o + 2, acc.z);
            unsafeAtomicAdd(o + 3, acc.w);
            acc = make_float4(0.f, 0.f, 0.f, 0.f);
            cur = d;
        }
        float v = vals[e];
        const float4 h = *(const float4*)(HW + src[e] * OUT_DIM + dbase);
        acc.x = fmaf(v, h.x, acc.x);
        acc.y = fmaf(v, h.y, acc.y);
        acc.z = fmaf(v, h.z, acc.z);
        acc.w = fmaf(v, h.w, acc.w);
    }
    float* o = out + cur * OUT_DIM + dbase;
    unsafeAtomicAdd(o + 0, acc.x);
    unsafeAtomicAdd(o + 1, acc.y);
    unsafeAtomicAdd(o + 2, acc.z);
    unsafeAtomicAdd(o + 3, acc.w);
}

// ---------------------------------------------------------------------------
extern "C" void kernel_launch(void* const* d_in, const int* in_sizes, int n_in,
                              void* d_out, int out_size, void* d_ws, size_t ws_size,
                              hipStream_t stream) {
    const float* H    = (const float*)d_in[0];
    const float* W    = (const float*)d_in[1];
    const float* bias = (const float*)d_in[2];
    const float* vals = (const float*)d_in[3];
    const int*   src  = (const int*)d_in[4];
    const int*   dst  = (const int*)d_in[5];
    float*       out  = (float*)d_out;

    const int n_nodes = in_sizes[0] / IN_DIM;   // 100000 (multiple of 16)
    const int E       = in_sizes[3];            // 1600000
    float* HWbuf      = (float*)d_ws;           // needs n_nodes*OUT_DIM*4 = 51.2 MB

    // 1) HW = H @ W  (WMMA)
    gemm_hw<<<n_nodes / 16, 256, 0, stream>>>(H, W, HWbuf);

    // 2) out = bias (broadcast)
    const int total = n_nodes * OUT_DIM;
    bias_init<<<(total + 255) / 256, 256, 0, stream>>>(bias, out, total);

    // 3) out += A @ HW  (segmented scatter over sorted dst)
    const int waves  = (E + EDGES_PER_WAVE - 1) / EDGES_PER_WAVE;
    const int blocks = (waves * 32 + 255) / 256;
    spmm_seg<<<blocks, 256, 0, stream>>>(HWbuf, vals, src, dst, out, E);
}